// GraphSAGE_13219909337542
// MI455X (gfx1250) — compile-verified
//
#include <hip/hip_runtime.h>

namespace {
constexpr int kNUser  = 100000;
constexpr int kNMovie = 50000;
constexpr int kNEdge  = 2000000;
constexpr int kNLbl   = 500000;
}

typedef float v2f __attribute__((ext_vector_type(2)));
typedef float v8f __attribute__((ext_vector_type(8)));

__device__ __forceinline__ v8f v8f_zero() {
  v8f z;
#pragma unroll
  for (int j = 0; j < 8; ++j) z[j] = 0.0f;
  return z;
}

// D = A(16x4 f32) * B(4x16 f32) + C(16x16 f32)
__device__ __forceinline__ v8f wmma4(v2f a, v2f b, v8f c) {
  return __builtin_amdgcn_wmma_f32_16x16x4_f32(false, a, false, b, (short)0, c,
                                               false, false);
}

__global__ __launch_bounds__(256) void k_zero(float4* __restrict__ p, int n4) {
  int i = blockIdx.x * 256 + threadIdx.x;
  if (i < n4) {
    float4 z; z.x = 0.f; z.y = 0.f; z.z = 0.f; z.w = 0.f;
    p[i] = z;
  }
}

__global__ __launch_bounds__(256) void k_count(const int* __restrict__ idx,
                                               float* __restrict__ cnt, int n) {
  int i = blockIdx.x * 256 + threadIdx.x;
  if (i < n) atomicAdd(&cnt[idx[i]], 1.0f);
}

__global__ __launch_bounds__(256) void k_inv(float* __restrict__ cnt, int n) {
  int i = blockIdx.x * 256 + threadIdx.x;
  if (i < n) cnt[i] = 1.0f / fmaxf(cnt[i], 1.0f);
}

// One wave (32 lanes) per edge: lane-coalesced read of a 64-float source row,
// float atomics into the destination's accumulation row (L2-resident).
__global__ __launch_bounds__(256) void k_scatter(const float* __restrict__ xsrc,
    const int* __restrict__ sidx, const int* __restrict__ didx,
    float* __restrict__ agg, int ne) {
  long tid = (long)blockIdx.x * 256 + threadIdx.x;
  int e = (int)(tid >> 5);
  if (e >= ne) return;
  int lane = (int)(tid & 31);
  long s = sidx[e];
  long d = didx[e];
  float v0 = xsrc[s * 64 + lane];
  float v1 = xsrc[s * 64 + 32 + lane];
  atomicAdd(&agg[d * 64 + lane], v0);
  atomicAdd(&agg[d * 64 + 32 + lane], v1);
}

// out[n,64] = maybe_relu( (agg*inv) @ Wl[64,64] + bl + xdst @ Wr[64,64] )
// One wave per 16-row tile; 4 output N-tiles via v_wmma_f32_16x16x4_f32.
__global__ __launch_bounds__(32) void k_sage(const float* __restrict__ agg,
    const float* __restrict__ inv, const float* __restrict__ xdst,
    const float* __restrict__ Wl, const float* __restrict__ bl,
    const float* __restrict__ Wr, float* __restrict__ out, int n, int relu) {
  int row0 = blockIdx.x * 16;
  if (row0 >= n) return;
  int lane = (int)threadIdx.x;
  int m = lane & 15;           // A: row / B: col / D: col
  int kh = (lane >> 4) << 1;   // K-half offset: 0 or 2
  int half = lane >> 4;
  long arow = (long)(row0 + m) * 64;
  float sc = inv[row0 + m];

  v8f acc[4];
#pragma unroll
  for (int nt = 0; nt < 4; ++nt) acc[nt] = v8f_zero();

  // mean @ Wl
#pragma unroll 4
  for (int ks = 0; ks < 16; ++ks) {
    int k0 = ks * 4;
    v2f a;
    a.x = agg[arow + k0 + kh] * sc;
    a.y = agg[arow + k0 + kh + 1] * sc;
#pragma unroll
    for (int nt = 0; nt < 4; ++nt) {
      v2f b;
      b.x = Wl[(k0 + kh) * 64 + nt * 16 + m];
      b.y = Wl[(k0 + kh + 1) * 64 + nt * 16 + m];
      acc[nt] = wmma4(a, b, acc[nt]);
    }
  }
  // x_dst @ Wr
#pragma unroll 4
  for (int ks = 0; ks < 16; ++ks) {
    int k0 = ks * 4;
    v2f a;
    a.x = xdst[arow + k0 + kh];
    a.y = xdst[arow + k0 + kh + 1];
#pragma unroll
    for (int nt = 0; nt < 4; ++nt) {
      v2f b;
      b.x = Wr[(k0 + kh) * 64 + nt * 16 + m];
      b.y = Wr[(k0 + kh + 1) * 64 + nt * 16 + m];
      acc[nt] = wmma4(a, b, acc[nt]);
    }
  }
  // epilogue: bias (+relu), D layout: VGPR q -> row q + 8*half, col = lane%16
#pragma unroll
  for (int nt = 0; nt < 4; ++nt) {
    float bias = bl[nt * 16 + m];
#pragma unroll
    for (int q = 0; q < 8; ++q) {
      float val = acc[nt][q] + bias;
      if (relu) val = fmaxf(val, 0.f);
      out[(long)(row0 + q + 8 * half) * 64 + nt * 16 + m] = val;
    }
  }
}

// Predictor: one wave per 16 label-edges. Gather concat rows into LDS
// (row stride 132 -> conflict-free column reads), then 3 WMMA GEMM layers.
__global__ __launch_bounds__(32) void k_pred(const float* __restrict__ xu,
    const float* __restrict__ xm, const int* __restrict__ lu,
    const int* __restrict__ lm, const float* __restrict__ W1,
    const float* __restrict__ b1, const float* __restrict__ W2,
    const float* __restrict__ b2, const float* __restrict__ W3,
    const float* __restrict__ b3, float* __restrict__ out, int nl) {
  constexpr int PAD = 132;
  __shared__ float h[16][PAD];
  __shared__ float t[16][PAD];
  int row0 = blockIdx.x * 16;
  if (row0 >= nl) return;
  int lane = (int)threadIdx.x;
  int m = lane & 15;
  int kh = (lane >> 4) << 1;
  int half = lane >> 4;

  // gather [xu[lu] | xm[lm]] -> h[16][128]
  for (int r = 0; r < 16; ++r) {
    long u = lu[row0 + r];
    long w = lm[row0 + r];
    h[r][lane]      = xu[u * 64 + lane];
    h[r][32 + lane] = xu[u * 64 + 32 + lane];
    h[r][64 + lane] = xm[w * 64 + lane];
    h[r][96 + lane] = xm[w * 64 + 32 + lane];
  }
  __syncthreads();

  // fc1: t = relu(h @ W1 + b1)   [16,128]x[128,128]
#pragma unroll
  for (int nt = 0; nt < 8; ++nt) {
    v8f acc = v8f_zero();
#pragma unroll 4
    for (int ks = 0; ks < 32; ++ks) {
      int k0 = ks * 4;
      v2f a; a.x = h[m][k0 + kh]; a.y = h[m][k0 + kh + 1];
      v2f b;
      b.x = W1[(k0 + kh) * 128 + nt * 16 + m];
      b.y = W1[(k0 + kh + 1) * 128 + nt * 16 + m];
      acc = wmma4(a, b, acc);
    }
    float bias = b1[nt * 16 + m];
#pragma unroll
    for (int q = 0; q < 8; ++q)
      t[q + 8 * half][nt * 16 + m] = fmaxf(acc[q] + bias, 0.f);
  }
  __syncthreads();

  // fc2: h = relu(t @ W2 + b2)
#pragma unroll
  for (int nt = 0; nt < 8; ++nt) {
    v8f acc = v8f_zero();
#pragma unroll 4
    for (int ks = 0; ks < 32; ++ks) {
      int k0 = ks * 4;
      v2f a; a.x = t[m][k0 + kh]; a.y = t[m][k0 + kh + 1];
      v2f b;
      b.x = W2[(k0 + kh) * 128 + nt * 16 + m];
      b.y = W2[(k0 + kh + 1) * 128 + nt * 16 + m];
      acc = wmma4(a, b, acc);
    }
    float bias = b2[nt * 16 + m];
#pragma unroll
    for (int q = 0; q < 8; ++q)
      h[q + 8 * half][nt * 16 + m] = fmaxf(acc[q] + bias, 0.f);
  }
  __syncthreads();

  // fc3: out = h @ W3 + b3   [16,128]x[128,5], N padded to 16 with zero cols
  {
    float msk = (m < 5) ? 1.f : 0.f;
    int cc = (m < 5) ? m : 0;
    v8f acc = v8f_zero();
#pragma unroll 4
    for (int ks = 0; ks < 32; ++ks) {
      int k0 = ks * 4;
      v2f a; a.x = h[m][k0 + kh]; a.y = h[m][k0 + kh + 1];
      v2f b;
      b.x = W3[(k0 + kh) * 5 + cc] * msk;
      b.y = W3[(k0 + kh + 1) * 5 + cc] * msk;
      acc = wmma4(a, b, acc);
    }
    if (m < 5) {
      float bias = b3[m];
#pragma unroll
      for (int q = 0; q < 8; ++q)
        out[(long)(row0 + q + 8 * half) * 5 + m] = acc[q] + bias;
    }
  }
}

extern "C" void kernel_launch(void* const* d_in, const int* in_sizes, int n_in,
                              void* d_out, int out_size, void* d_ws, size_t ws_size,
                              hipStream_t stream) {
  (void)in_sizes; (void)n_in; (void)out_size; (void)ws_size;
  const float* x_user  = (const float*)d_in[0];
  const float* x_movie = (const float*)d_in[1];
  // d_in[2] = edge_weight (unused by vanilla SAGEConv, per reference)
  const int* e_src = (const int*)d_in[3];
  const int* e_dst = (const int*)d_in[4];
  const int* l_usr = (const int*)d_in[5];
  const int* l_mov = (const int*)d_in[6];
  const float *Wl[3][2], *bl[3][2], *Wr[3][2];
  for (int i = 0; i < 3; ++i)
    for (int et = 0; et < 2; ++et) {  // et 0 = um, 1 = mu (dict order)
      int base = 7 + i * 6 + et * 3;
      Wl[i][et] = (const float*)d_in[base];
      bl[i][et] = (const float*)d_in[base + 1];
      Wr[i][et] = (const float*)d_in[base + 2];
    }
  const float* W1 = (const float*)d_in[25];
  const float* b1 = (const float*)d_in[26];
  const float* W2 = (const float*)d_in[27];
  const float* b2 = (const float*)d_in[28];
  const float* W3 = (const float*)d_in[29];
  const float* b3 = (const float*)d_in[30];

  float* p = (float*)d_ws;
  float* xu_a = p;  p += (long)kNUser * 64;
  float* xm_a = p;  p += (long)kNMovie * 64;
  float* xu_b = p;  p += (long)kNUser * 64;
  float* xm_b = p;  p += (long)kNMovie * 64;
  float* agg_u = p; p += (long)kNUser * 64;
  float* agg_m = p; p += (long)kNMovie * 64;
  float* inv_u = p; p += kNUser;
  float* inv_m = p; p += kNMovie;

  auto zero = [&](float* q, long nfl) {
    int n4 = (int)(nfl / 4);
    k_zero<<<(n4 + 255) / 256, 256, 0, stream>>>((float4*)q, n4);
  };

  // degrees -> reciprocal of max(cnt,1)
  zero(inv_u, kNUser);
  zero(inv_m, kNMovie);
  k_count<<<(kNEdge + 255) / 256, 256, 0, stream>>>(e_src, inv_u, kNEdge);
  k_count<<<(kNEdge + 255) / 256, 256, 0, stream>>>(e_dst, inv_m, kNEdge);
  k_inv<<<(kNUser + 255) / 256, 256, 0, stream>>>(inv_u, kNUser);
  k_inv<<<(kNMovie + 255) / 256, 256, 0, stream>>>(inv_m, kNMovie);

  const float* cu = x_user;
  const float* cm = x_movie;
  float* nu_[3] = {xu_a, xu_b, xu_a};
  float* nm_[3] = {xm_a, xm_b, xm_a};
  int scatterBlocks = (int)(((long)kNEdge * 32 + 255) / 256);

  for (int i = 0; i < 3; ++i) {
    zero(agg_m, (long)kNMovie * 64);
    zero(agg_u, (long)kNUser * 64);
    // um: user -> movie messages ; mu: movie -> user messages
    k_scatter<<<scatterBlocks, 256, 0, stream>>>(cu, e_src, e_dst, agg_m, kNEdge);
    k_scatter<<<scatterBlocks, 256, 0, stream>>>(cm, e_dst, e_src, agg_u, kNEdge);
    int relu = (i < 2) ? 1 : 0;
    k_sage<<<kNMovie / 16, 32, 0, stream>>>(agg_m, inv_m, cm, Wl[i][0], bl[i][0],
                                            Wr[i][0], nm_[i], kNMovie, relu);
    k_sage<<<kNUser / 16, 32, 0, stream>>>(agg_u, inv_u, cu, Wl[i][1], bl[i][1],
                                           Wr[i][1], nu_[i], kNUser, relu);
    cu = nu_[i];
    cm = nm_[i];
  }

  k_pred<<<kNLbl / 16, 32, 0, stream>>>(cu, cm, l_usr, l_mov, W1, b1, W2, b2,
                                        W3, b3, (float*)d_out, kNLbl);
}